// TripletCenterCosineRFLoss_15917148799611
// MI455X (gfx1250) — compile-verified
//
#include <hip/hip_runtime.h>
#include <math.h>

// Problem constants (match reference).
#define BB 8192
#define CC 1000
#define DD 1024
#define NEG_INF (-__builtin_inff())

#define MT 2          // 16-row A tiles per block in the sims kernel (32 rows)
#define KCH 8         // k-steps (of 4) per unrolled chunk -> 32 floats, 128 B

// sched_group_barrier masks (LLVM AMDGPU)
#define SG_WMMA      0x008   // MFMA/WMMA
#define SG_VMEM_READ 0x020   // VMEM reads

typedef float v2f __attribute__((ext_vector_type(2)));
typedef float v8f __attribute__((ext_vector_type(8)));

// Force global (addrspace 1) loads in the hot loops: generic->AS1 via integer
// round-trip (global and generic share address values on AMDGPU).
typedef const __attribute__((address_space(1))) v2f* gv2p;
__device__ __forceinline__ v2f gload2(const float* p) {
    return *(gv2p)(unsigned long long)p;
}

// Max across a 16-lane half of a wave32 (xor masks 1..8 stay within the half).
__device__ __forceinline__ float half16_max(float v) {
    v = fmaxf(v, __shfl_xor(v, 1, 32));
    v = fmaxf(v, __shfl_xor(v, 2, 32));
    v = fmaxf(v, __shfl_xor(v, 4, 32));
    v = fmaxf(v, __shfl_xor(v, 8, 32));
    return v;
}

__global__ void zero_out_kernel(float* out) {
    if (threadIdx.x == 0 && blockIdx.x == 0) out[0] = 0.0f;
}

// One block per center row: nc[r] = centers[r] / ||centers[r]||2
__global__ void norm_centers_kernel(const float* __restrict__ centers,
                                    float* __restrict__ nc) {
    const int row = blockIdx.x;
    const float* src = centers + (size_t)row * DD;
    float ss = 0.0f;
    for (int d = threadIdx.x; d < DD; d += 256) {
        float v = src[d];
        ss += v * v;
    }
    for (int m = 1; m < 32; m <<= 1) ss += __shfl_xor(ss, m, 32);
    __shared__ float sred[8];
    if ((threadIdx.x & 31) == 0) sred[threadIdx.x >> 5] = ss;
    __syncthreads();
    float tot = 0.0f;
#pragma unroll
    for (int w = 0; w < 8; ++w) tot += sred[w];
    const float inv = rsqrtf(tot);
    float* dst = nc + (size_t)row * DD;
    for (int d = threadIdx.x; d < DD; d += 256) dst[d] = src[d] * inv;
}

// ---------------------------------------------------------------------------
// WMMA f32 16x16x4 fragment layouts (ISA 7.12.2, wave32):
//   A (16x4, MxK): lane m=lane&15, khalf=lane>>4; VGPR0=K(2*khalf), VGPR1=K(2*khalf+1)
//   B (4x16, KxN): mirrored: lane n=lane&15, khalf=lane>>4
//   C/D (16x16):   VGPR v, lanes 0-15 -> M=v,N=lane; lanes 16-31 -> M=v+8,N=lane-16
// 8 waves x 8 column tiles cover 64*16 = 1024 padded columns (>= CC=1000).
// sched_group_barrier pins the per-chunk order to
//   VMEM(loads_j) ; [VMEM(loads_{j+1}), WMMA_j] ... ; WMMA_last
// giving a one-deep load/WMMA software pipeline with staggered waitcnts.
// ---------------------------------------------------------------------------

// c2c stage: rows = raw centers, cols = normalized centers, exclude diagonal.
__global__ void c2c_max_kernel(const float* __restrict__ centers,
                               const float* __restrict__ nc,
                               float* __restrict__ maxc2c) {
    const int lane  = threadIdx.x & 31;
    const int wave  = threadIdx.x >> 5;
    const int n     = lane & 15;
    const int khalf = lane >> 4;
    const int row0  = blockIdx.x * 16;

    int arow = row0 + n;
    if (arow >= CC) arow = CC - 1;   // clamp; masked later
    const float* ap = centers + (size_t)arow * DD + 2 * khalf;

    const v8f vzero = {0.f, 0.f, 0.f, 0.f, 0.f, 0.f, 0.f, 0.f};
    v8f acc[8];
    const float* bp[8];
#pragma unroll
    for (int t = 0; t < 8; ++t) {
        acc[t] = vzero;
        int bcol = (wave * 8 + t) * 16 + n;
        if (bcol >= CC) bcol = CC - 1;
        bp[t] = nc + (size_t)bcol * DD + 2 * khalf;
    }

    for (int chunk = 0; chunk < DD / (4 * KCH); ++chunk) {
#pragma unroll
        for (int j = 0; j < KCH; ++j) {
            const v2f a = gload2(ap + 4 * j);
            v2f b[8];
#pragma unroll
            for (int t = 0; t < 8; ++t) b[t] = gload2(bp[t] + 4 * j);
#pragma unroll
            for (int t = 0; t < 8; ++t) {
                acc[t] = __builtin_amdgcn_wmma_f32_16x16x4_f32(
                    false, a, false, b[t], (short)0, acc[t], false, false);
            }
        }
        // Pipeline order: loads(j) ahead of wmma(j-1).
        __builtin_amdgcn_sched_group_barrier(SG_VMEM_READ, 9, 0);
#pragma unroll
        for (int j = 1; j < KCH; ++j) {
            __builtin_amdgcn_sched_group_barrier(SG_VMEM_READ, 9, 0);
            __builtin_amdgcn_sched_group_barrier(SG_WMMA, 8, 0);
        }
        __builtin_amdgcn_sched_group_barrier(SG_WMMA, 8, 0);

        ap += 4 * KCH;
#pragma unroll
        for (int t = 0; t < 8; ++t) bp[t] += 4 * KCH;
    }

    float rowmax[8];
#pragma unroll
    for (int v = 0; v < 8; ++v) rowmax[v] = NEG_INF;
#pragma unroll
    for (int t = 0; t < 8; ++t) {
        const int col = (wave * 8 + t) * 16 + n;
#pragma unroll
        for (int v = 0; v < 8; ++v) {
            const int row = row0 + v + 8 * khalf;
            const bool ok = (col < CC) && (col != row);
            rowmax[v] = fmaxf(rowmax[v], ok ? acc[t][v] : NEG_INF);
        }
    }

    __shared__ float red[8][16];
#pragma unroll
    for (int v = 0; v < 8; ++v) {
        const float r = half16_max(rowmax[v]);
        if (n == 0) red[wave][v + 8 * khalf] = r;
    }
    __syncthreads();

    if (threadIdx.x < 16) {
        const int row = row0 + threadIdx.x;
        if (row < CC) {
            float m = NEG_INF;
#pragma unroll
            for (int w = 0; w < 8; ++w) m = fmaxf(m, red[w][threadIdx.x]);
            maxc2c[row] = m;
        }
    }
}

// sims stage: 32 rows of x per block (two 16-row A tiles sharing every B
// fragment -> halves nc L2 traffic). Tracks own-class sim and max over the
// rest, folds in the c2c term, atomically accumulates the loss partial.
__global__ void sims_loss_kernel(const float* __restrict__ x,
                                 const int* __restrict__ labels,
                                 const float* __restrict__ nc,
                                 const float* __restrict__ maxc2c,
                                 float* __restrict__ out) {
    const int lane  = threadIdx.x & 31;
    const int wave  = threadIdx.x >> 5;
    const int n     = lane & 15;
    const int khalf = lane >> 4;
    const int row0  = blockIdx.x * (16 * MT);

    __shared__ int slab[16 * MT];
    if (threadIdx.x < 16 * MT) slab[threadIdx.x] = labels[row0 + threadIdx.x];
    __syncthreads();

    const float* ap0 = x + (size_t)(row0 + n) * DD + 2 * khalf;
    const float* ap1 = ap0 + (size_t)16 * DD;

    const v8f vzero = {0.f, 0.f, 0.f, 0.f, 0.f, 0.f, 0.f, 0.f};
    v8f acc[MT][8];
    const float* bp[8];
#pragma unroll
    for (int t = 0; t < 8; ++t) {
#pragma unroll
        for (int r = 0; r < MT; ++r) acc[r][t] = vzero;
        int bcol = (wave * 8 + t) * 16 + n;
        if (bcol >= CC) bcol = CC - 1;
        bp[t] = nc + (size_t)bcol * DD + 2 * khalf;
    }

    for (int chunk = 0; chunk < DD / (4 * KCH); ++chunk) {
#pragma unroll
        for (int j = 0; j < KCH; ++j) {
            const v2f a0 = gload2(ap0 + 4 * j);
            const v2f a1 = gload2(ap1 + 4 * j);
            v2f b[8];
#pragma unroll
            for (int t = 0; t < 8; ++t) b[t] = gload2(bp[t] + 4 * j);
#pragma unroll
            for (int t = 0; t < 8; ++t) {
                acc[0][t] = __builtin_amdgcn_wmma_f32_16x16x4_f32(
                    false, a0, false, b[t], (short)0, acc[0][t], false, false);
                acc[1][t] = __builtin_amdgcn_wmma_f32_16x16x4_f32(
                    false, a1, false, b[t], (short)0, acc[1][t], false, false);
            }
        }
        // Pipeline order: loads(j) ahead of wmma(j-1).
        __builtin_amdgcn_sched_group_barrier(SG_VMEM_READ, 10, 0);
#pragma unroll
        for (int j = 1; j < KCH; ++j) {
            __builtin_amdgcn_sched_group_barrier(SG_VMEM_READ, 10, 0);
            __builtin_amdgcn_sched_group_barrier(SG_WMMA, 16, 0);
        }
        __builtin_amdgcn_sched_group_barrier(SG_WMMA, 16, 0);

        ap0 += 4 * KCH;
        ap1 += 4 * KCH;
#pragma unroll
        for (int t = 0; t < 8; ++t) bp[t] += 4 * KCH;
    }

    float rowmax[MT][8], rowown[MT][8];
#pragma unroll
    for (int r = 0; r < MT; ++r)
#pragma unroll
        for (int v = 0; v < 8; ++v) { rowmax[r][v] = NEG_INF; rowown[r][v] = NEG_INF; }

#pragma unroll
    for (int t = 0; t < 8; ++t) {
        const int col = (wave * 8 + t) * 16 + n;
        const bool incol = (col < CC);
#pragma unroll
        for (int r = 0; r < MT; ++r) {
#pragma unroll
            for (int v = 0; v < 8; ++v) {
                const int rloc = 16 * r + v + 8 * khalf;
                const int lab  = slab[rloc];
                const float s  = acc[r][t][v];
                const bool own = incol && (col == lab);
                rowmax[r][v] = fmaxf(rowmax[r][v], (incol && !own) ? s : NEG_INF);
                rowown[r][v] = fmaxf(rowown[r][v], own ? s : NEG_INF);
            }
        }
    }

    __shared__ float redmax[8][16 * MT];
    __shared__ float redown[8][16 * MT];
#pragma unroll
    for (int r = 0; r < MT; ++r) {
#pragma unroll
        for (int v = 0; v < 8; ++v) {
            const float rm = half16_max(rowmax[r][v]);
            const float ro = half16_max(rowown[r][v]);
            if (n == 0) {
                redmax[wave][16 * r + v + 8 * khalf] = rm;
                redown[wave][16 * r + v + 8 * khalf] = ro;
            }
        }
    }
    __syncthreads();

    __shared__ float partial[16 * MT];
    if (threadIdx.x < 16 * MT) {
        float m = NEG_INF, o = NEG_INF;
#pragma unroll
        for (int w = 0; w < 8; ++w) {
            m = fmaxf(m, redmax[w][threadIdx.x]);
            o = fmaxf(o, redown[w][threadIdx.x]);
        }
        const int lab = slab[threadIdx.x];
        // pos = 1 - o ; neg_center = 1 - m ; term1 = relu(pos + 1.0 - neg)
        const float term1 = fmaxf(0.0f, (1.0f - o) + 1.0f - (1.0f - m));
        // neg_c2c = 1 - maxc2c[lab] ; term2 = relu(1.4 - neg_c2c)
        const float term2 = fmaxf(0.0f, 1.4f - (1.0f - maxc2c[lab]));
        partial[threadIdx.x] = term1 + term2;
    }
    __syncthreads();

    if (threadIdx.x == 0) {
        float s = 0.0f;
#pragma unroll
        for (int i = 0; i < 16 * MT; ++i) s += partial[i];
        atomicAdd(out, s * (1.0f / (float)BB));
    }
}

extern "C" void kernel_launch(void* const* d_in, const int* in_sizes, int n_in,
                              void* d_out, int out_size, void* d_ws, size_t ws_size,
                              hipStream_t stream) {
    (void)in_sizes; (void)n_in; (void)out_size; (void)ws_size;
    const float* x       = (const float*)d_in[0];
    const int*   labels  = (const int*)d_in[1];
    const float* centers = (const float*)d_in[2];
    float*       out     = (float*)d_out;

    float* nc     = (float*)d_ws;          // CC*DD floats  (4 MB)
    float* maxc2c = nc + (size_t)CC * DD;  // CC floats

    zero_out_kernel<<<1, 64, 0, stream>>>(out);
    norm_centers_kernel<<<CC, 256, 0, stream>>>(centers, nc);
    c2c_max_kernel<<<(CC + 15) / 16, 256, 0, stream>>>(centers, nc, maxc2c);
    sims_loss_kernel<<<BB / (16 * MT), 256, 0, stream>>>(x, labels, nc, maxc2c, out);
}